// MAE_52613349376068
// MI455X (gfx1250) — compile-verified
//
#include <hip/hip_runtime.h>
#include <cstdint>

#define WAVE            32
#define WAVES_PER_BLOCK 8
#define BLOCK_THREADS   (WAVE * WAVES_PER_BLOCK)

// ---------------------------------------------------------------------------
// Wave32 butterfly reductions (CDNA5 is wave32-only).
// ---------------------------------------------------------------------------
__device__ __forceinline__ float wave_max(float m) {
    #pragma unroll
    for (int off = 16; off > 0; off >>= 1)
        m = fmaxf(m, __shfl_xor(m, off, WAVE));
    return m;
}
__device__ __forceinline__ float wave_sum(float s) {
    #pragma unroll
    for (int off = 16; off > 0; off >>= 1)
        s += __shfl_xor(s, off, WAVE);
    return s;
}

// ---------------------------------------------------------------------------
// Issue one row (NV float4s) as async global->LDS b128 copies.
// Exactly ceil(NV/32) async instructions per wave (partial EXEC on last one
// still issues and increments ASYNCcnt) -- this count is relied on by the
// partial s_wait_asynccnt in the pipelined kernel.
// ---------------------------------------------------------------------------
template<int NV>
__device__ __forceinline__ void issue_row_async(uint64_t gbase, uint32_t lbase, int lane) {
    #pragma unroll
    for (int k = 0; k < (NV + WAVE - 1) / WAVE; ++k) {
        const int i = lane + k * WAVE;
        if (i < NV) {
            uint64_t ga = gbase + (uint64_t)i * 16u;
            uint32_t la = lbase + (uint32_t)i * 16u;
            asm volatile("global_load_async_to_lds_b128 %0, %1, off"
                         :: "v"(la), "v"(ga) : "memory");
        }
    }
}

// ---------------------------------------------------------------------------
// Two-pass stable softmax over one LDS-staged row; returns p_target
// (same value in all lanes). Fully unrolled for compile-time C (C % 4 == 0).
// ---------------------------------------------------------------------------
template<int C>
__device__ __forceinline__ float row_softmax_p(const float* wbuf, int lane, int tgt) {
    constexpr int NV = C / 4;
    const float4* vb = reinterpret_cast<const float4*>(wbuf);

    float m = -3.402823466e38f;
    #pragma unroll
    for (int k = 0; k < (NV + WAVE - 1) / WAVE; ++k) {
        const int i = lane + k * WAVE;
        if (i < NV) {
            float4 v = vb[i];
            m = fmaxf(m, fmaxf(fmaxf(v.x, v.y), fmaxf(v.z, v.w)));
        }
    }
    m = wave_max(m);

    const float L2E = 1.4426950408889634f;
    float s = 0.0f;
    #pragma unroll
    for (int k = 0; k < (NV + WAVE - 1) / WAVE; ++k) {
        const int i = lane + k * WAVE;
        if (i < NV) {
            float4 v = vb[i];
            s += __builtin_amdgcn_exp2f((v.x - m) * L2E);
            s += __builtin_amdgcn_exp2f((v.y - m) * L2E);
            s += __builtin_amdgcn_exp2f((v.z - m) * L2E);
            s += __builtin_amdgcn_exp2f((v.w - m) * L2E);
        }
    }
    s = wave_sum(s);

    const float xt = wbuf[tgt];   // broadcast LDS read
    return __builtin_amdgcn_exp2f((xt - m) * L2E) / s;
}

// ---------------------------------------------------------------------------
// Specialized pipelined kernel: 2 rows per wave, double-buffered in LDS.
// Issue row0 + row1 asyncs, partial-wait (asynccnt <= ITERS) so row0 is
// complete while row1 streams from HBM under the compute of row0.
// ---------------------------------------------------------------------------
template<int C>
__global__ __launch_bounds__(BLOCK_THREADS)
void focal_rows_pipelined(const float* __restrict__ outputs,
                          const long long* __restrict__ targets,
                          float* __restrict__ partials, int B)
{
    extern __shared__ float sdata[];
    constexpr int RF    = ((C + 3) & ~3) + 4;          // padded row (floats), 16B-aligned
    constexpr int NV    = C / 4;
    constexpr int ITERS = (NV + WAVE - 1) / WAVE;      // async instrs per row (=8 for C=1000)

    const int lane = threadIdx.x & (WAVE - 1);
    const int wave = threadIdx.x >> 5;
    const int pair = blockIdx.x * WAVES_PER_BLOCK + wave;
    const int row0 = pair * 2;
    const int row1 = row0 + 1;

    float* buf0 = sdata + (size_t)wave * (2 * RF);
    float* buf1 = buf0 + RF;

    const bool do0 = row0 < B;
    const bool do1 = row1 < B;

    if (do0) issue_row_async<NV>((uint64_t)(uintptr_t)(outputs + (size_t)row0 * C),
                                 (uint32_t)(uintptr_t)buf0, lane);
    if (do1) issue_row_async<NV>((uint64_t)(uintptr_t)(outputs + (size_t)row1 * C),
                                 (uint32_t)(uintptr_t)buf1, lane);

    float psum = 0.0f;
    if (do0) {
        if (do1) asm volatile("s_wait_asynccnt %0" :: "i"(ITERS) : "memory"); // row0 done, row1 in flight
        else     asm volatile("s_wait_asynccnt 0" ::: "memory");
        psum += row_softmax_p<C>(buf0, lane, (int)targets[row0]);
    }
    if (do1) {
        asm volatile("s_wait_asynccnt 0" ::: "memory");
        psum += row_softmax_p<C>(buf1, lane, (int)targets[row1]);
    }

    // deterministic per-block aggregation
    float* pshare = sdata + (size_t)WAVES_PER_BLOCK * (2 * RF);
    if (lane == 0) pshare[wave] = psum;
    __syncthreads();
    if (threadIdx.x == 0) {
        float acc = 0.0f;
        #pragma unroll
        for (int w = 0; w < WAVES_PER_BLOCK; ++w) acc += pshare[w];
        partials[blockIdx.x] = acc;
    }
}

// ---------------------------------------------------------------------------
// Generic fallback (runtime C, one row per wave, non-pipelined).
// ---------------------------------------------------------------------------
__global__ __launch_bounds__(BLOCK_THREADS)
void focal_rows_generic(const float* __restrict__ outputs,
                        const long long* __restrict__ targets,
                        float* __restrict__ partials,
                        int B, int C, int rowFloats)
{
    extern __shared__ float sdata[];
    const int lane = threadIdx.x & (WAVE - 1);
    const int wave = threadIdx.x >> 5;
    const int row  = blockIdx.x * WAVES_PER_BLOCK + wave;

    float p = 0.0f;
    if (row < B) {
        float* wbuf = sdata + (size_t)wave * rowFloats;
        const int nVec = C >> 2;
        const uint64_t gbase = (uint64_t)(uintptr_t)(outputs + (size_t)row * C);
        const uint32_t lbase = (uint32_t)(uintptr_t)wbuf;

        for (int i = lane; i < nVec; i += WAVE) {
            uint64_t ga = gbase + (uint64_t)i * 16u;
            uint32_t la = lbase + (uint32_t)i * 16u;
            asm volatile("global_load_async_to_lds_b128 %0, %1, off"
                         :: "v"(la), "v"(ga) : "memory");
        }
        for (int j = (nVec << 2) + lane; j < C; j += WAVE) {
            uint64_t ga = gbase + (uint64_t)j * 4u;
            uint32_t la = lbase + (uint32_t)j * 4u;
            asm volatile("global_load_async_to_lds_b32 %0, %1, off"
                         :: "v"(la), "v"(ga) : "memory");
        }
        asm volatile("s_wait_asynccnt 0" ::: "memory");

        float m = -3.402823466e38f;
        for (int i = lane; i < nVec; i += WAVE) {
            float4 v = reinterpret_cast<const float4*>(wbuf)[i];
            m = fmaxf(m, fmaxf(fmaxf(v.x, v.y), fmaxf(v.z, v.w)));
        }
        for (int j = (nVec << 2) + lane; j < C; j += WAVE) m = fmaxf(m, wbuf[j]);
        m = wave_max(m);

        const float L2E = 1.4426950408889634f;
        float s = 0.0f;
        for (int i = lane; i < nVec; i += WAVE) {
            float4 v = reinterpret_cast<const float4*>(wbuf)[i];
            s += __builtin_amdgcn_exp2f((v.x - m) * L2E);
            s += __builtin_amdgcn_exp2f((v.y - m) * L2E);
            s += __builtin_amdgcn_exp2f((v.z - m) * L2E);
            s += __builtin_amdgcn_exp2f((v.w - m) * L2E);
        }
        for (int j = (nVec << 2) + lane; j < C; j += WAVE)
            s += __builtin_amdgcn_exp2f((wbuf[j] - m) * L2E);
        s = wave_sum(s);

        if (lane == 0) {
            int t = (int)targets[row];
            p = __builtin_amdgcn_exp2f((wbuf[t] - m) * L2E) / s;
        }
    }

    float* pshare = sdata + (size_t)WAVES_PER_BLOCK * rowFloats;
    if (lane == 0) pshare[wave] = p;
    __syncthreads();
    if (threadIdx.x == 0) {
        float acc = 0.0f;
        #pragma unroll
        for (int w = 0; w < WAVES_PER_BLOCK; ++w) acc += pshare[w];
        partials[blockIdx.x] = acc;
    }
}

// ---------------------------------------------------------------------------
// Deterministic final reduction: out = 2 - 2 * (sum p_t) / B
// ---------------------------------------------------------------------------
__global__ __launch_bounds__(1024)
void focal_reduce_kernel(const float* __restrict__ partials, int n,
                         float* __restrict__ out, float invB)
{
    __shared__ float sh[1024];
    float s = 0.0f;
    for (int i = threadIdx.x; i < n; i += 1024)   // fixed order per thread
        s += partials[i];
    sh[threadIdx.x] = s;
    __syncthreads();
    for (int off = 512; off > 0; off >>= 1) {
        if ((int)threadIdx.x < off) sh[threadIdx.x] += sh[threadIdx.x + off];
        __syncthreads();
    }
    if (threadIdx.x == 0)
        out[0] = 2.0f - 2.0f * sh[0] * invB;
}

// ---------------------------------------------------------------------------
extern "C" void kernel_launch(void* const* d_in, const int* in_sizes, int n_in,
                              void* d_out, int out_size, void* d_ws, size_t ws_size,
                              hipStream_t stream)
{
    const float*     outputs = (const float*)d_in[0];     // [B, C] fp32
    const long long* targets = (const long long*)d_in[1]; // [B] int64
    float* out      = (float*)d_out;                      // scalar fp32
    float* partials = (float*)d_ws;

    const int B = in_sizes[1];        // 131072
    const int C = in_sizes[0] / B;    // 1000

    if (C == 1000) {
        constexpr int Cc = 1000;
        constexpr int RF = ((Cc + 3) & ~3) + 4;                      // 1004 floats
        const int rowsPerBlock = WAVES_PER_BLOCK * 2;                // 16
        const int blocks = (B + rowsPerBlock - 1) / rowsPerBlock;    // 8192
        const size_t shmem =
            ((size_t)WAVES_PER_BLOCK * 2 * RF + WAVES_PER_BLOCK) * sizeof(float); // ~64.3 KB

        focal_rows_pipelined<Cc><<<blocks, BLOCK_THREADS, shmem, stream>>>(
            outputs, targets, partials, B);
        focal_reduce_kernel<<<1, 1024, 0, stream>>>(
            partials, blocks, out, 1.0f / (float)B);
    } else {
        const int rowFloats = ((C + 3) & ~3) + 8;
        const int blocks = (B + WAVES_PER_BLOCK - 1) / WAVES_PER_BLOCK;
        const size_t shmem =
            ((size_t)WAVES_PER_BLOCK * rowFloats + WAVES_PER_BLOCK) * sizeof(float);

        focal_rows_generic<<<blocks, BLOCK_THREADS, shmem, stream>>>(
            outputs, targets, partials, B, C, rowFloats);
        focal_reduce_kernel<<<1, 1024, 0, stream>>>(
            partials, blocks, out, 1.0f / (float)B);
    }
}